// hess_gelu_15676630630556
// MI455X (gfx1250) — compile-verified
//
#include <hip/hip_runtime.h>
#include <hip/hip_bf16.h>

// hess_gelu: out = (2 - 2*x^2) * (1/sqrt(2*pi)) * exp(-x^2/2)
// HBM-bound: 2.15 GB traffic -> ~92 us floor at 23.3 TB/s. Compute (~6 flops +
// 1 v_exp_f32 per element) is negligible. Optimal path: branch-free 128-bit
// non-temporal loads/stores (stream >> 192 MB L2), hardware exp2.

typedef __attribute__((ext_vector_type(4))) float v4f;

#define INV_SQRT_2PI   0.3989422804014327f
#define NEG_HALF_LOG2E -0.7213475204444817f   // -0.5 * log2(e)

__device__ __forceinline__ float hess_gelu_scalar(float x) {
    float t = x * x;
    // exp(-0.5*t) == exp2(t * (-0.5*log2(e))) -> single v_exp_f32
    float e = __builtin_amdgcn_exp2f(t * NEG_HALF_LOG2E);
    return (2.0f - 2.0f * t) * (INV_SQRT_2PI * e);
}

__device__ __forceinline__ v4f hess_gelu_v4(v4f v) {
    v4f r;
    r.x = hess_gelu_scalar(v.x);
    r.y = hess_gelu_scalar(v.y);
    r.z = hess_gelu_scalar(v.z);
    r.w = hess_gelu_scalar(v.w);
    return r;
}

#define TPB 256
#define VPT 4   // float4 chunks per thread -> 64 B/thread, 4 B128 loads in flight

// Fast path: caller guarantees n4 % (TPB*VPT) == 0. No bounds checks ->
// no exec-mask branches; loads batch into one s_clause, single s_wait_loadcnt.
__global__ __launch_bounds__(TPB) void hess_gelu_fast(
    const v4f* __restrict__ x, v4f* __restrict__ out) {
    int base = blockIdx.x * (TPB * VPT) + threadIdx.x;

    v4f val[VPT];
    #pragma unroll
    for (int k = 0; k < VPT; ++k) {
        val[k] = __builtin_nontemporal_load(&x[base + k * TPB]);  // th:NT
    }
    #pragma unroll
    for (int k = 0; k < VPT; ++k) {
        __builtin_nontemporal_store(hess_gelu_v4(val[k]), &out[base + k * TPB]);
    }
}

// Generic fallback for sizes not divisible by TPB*VPT*4 floats (unused for
// the 16384^2 harness shape, but keeps kernel_launch correct for any size).
__global__ __launch_bounds__(TPB) void hess_gelu_tail(
    const float* __restrict__ x, float* __restrict__ out, int n) {
    int i = blockIdx.x * TPB + threadIdx.x;
    if (i < n) {
        out[i] = hess_gelu_scalar(x[i]);
    }
}

extern "C" void kernel_launch(void* const* d_in, const int* in_sizes, int n_in,
                              void* d_out, int out_size, void* d_ws, size_t ws_size,
                              hipStream_t stream) {
    int n = in_sizes[0];  // 268,435,456 for the harness shape

    const int elems_per_block = TPB * VPT * 4;  // floats per block = 4096

    if ((n % elems_per_block) == 0) {
        const v4f* x   = (const v4f*)d_in[0];
        v4f*       out = (v4f*)d_out;
        int grid = n / elems_per_block;          // 65536 blocks
        hess_gelu_fast<<<grid, TPB, 0, stream>>>(x, out);
    } else {
        const float* x   = (const float*)d_in[0];
        float*       out = (float*)d_out;
        int grid = (n + TPB - 1) / TPB;
        hess_gelu_tail<<<grid, TPB, 0, stream>>>(x, out, n);
    }
}